// T3A_73443940761871
// MI455X (gfx1250) — compile-verified
//
#include <hip/hip_runtime.h>

// ---------------- problem constants ----------------
#define C_CLS   1000
#define B_ROWS  1024
#define D_FEAT  2048
#define IN_FEAT 8192
#define N_SUP   (C_CLS + B_ROWS)   // 2024 supports
#define FILTK   100

typedef __attribute__((ext_vector_type(2))) float v2f_t;
typedef __attribute__((ext_vector_type(8))) float v8f_t;

// -------- CDNA5 async global->LDS copy (ASYNCcnt-tracked), with sync fallback --------
#if defined(__HIP_DEVICE_COMPILE__) && __has_builtin(__builtin_amdgcn_global_load_async_to_lds_b128)
#define HAVE_ASYNC_LDS 1
#else
#define HAVE_ASYNC_LDS 0
#endif

#if HAVE_ASYNC_LDS
typedef int v4i_vs __attribute__((vector_size(16)));
typedef __attribute__((address_space(1))) v4i_vs* g4p_t;   // global int4*
typedef __attribute__((address_space(3))) v4i_vs* l4p_t;   // LDS int4*
#endif

__device__ __forceinline__ void async_copy_b128(const float* g, float* l)
{
#if HAVE_ASYNC_LDS
    __builtin_amdgcn_global_load_async_to_lds_b128(
        (g4p_t)(const void*)g, (l4p_t)(void*)l, 0, 0);
#else
    (void)g; (void)l;
#endif
}

__device__ __forceinline__ void wait_async_all()
{
#if HAVE_ASYNC_LDS
#if __has_builtin(__builtin_amdgcn_s_wait_asynccnt)
    __builtin_amdgcn_s_wait_asynccnt(0);
#else
    asm volatile("s_wait_asynccnt 0x0" ::: "memory");
#endif
#endif
}

// ---------------- WMMA f32 GEMM (double-buffered LDS, async staging) ----------------
// C[M,N] = A[M,K] * op(B) (+ epilogue), all f32.
// BT=true : op(B)[k][n] = B[n*ldb + k]   (B stored [N,K] row-major)
// BT=false: op(B)[k][n] = B[k*ldb + n]   (B stored [K,N] row-major)
// mode 0: C = acc + bias[n]
// mode 1: C = 0.5*acc*rscale[n] + 0.5*P[m*ldp+n] + 0.5*bias[n]
#define BM   128
#define BN   128
#define KT   16
#define LDST 20   // padded LDS stride (floats): 16B-aligned rows, conflict-free frag loads

template<bool BT>
__global__ __launch_bounds__(256)
void gemm_f32_wmma(const float* __restrict__ A, int lda,
                   const float* __restrict__ B, int ldb,
                   float* __restrict__ C, int ldc,
                   int M, int N, int K, int Nb,
                   const float* __restrict__ bias, int mode,
                   const float* __restrict__ rscale,
                   const float* __restrict__ P, int ldp)
{
    __shared__ __align__(16) float Al[2][BM * LDST];
    __shared__ __align__(16) float Bl[2][BN * LDST];

    const int tid  = threadIdx.x;
    const int w    = tid >> 5;       // wave id 0..7
    const int lane = tid & 31;
    const int lh   = lane >> 4;      // lane half (0/1)
    const int l16  = lane & 15;
    const int wm   = w >> 1;         // 0..3 : 32-row strip
    const int wn   = w & 1;          // 0..1 : 64-col strip
    const int m0   = blockIdx.y * BM;
    const int n0   = blockIdx.x * BN;
    const int ntiles = K / KT;

    // ---- tile staging (A always; B per layout) into buffer `buf` for k-slab k0 ----
    auto stage = [&](int buf, int k0) {
#pragma unroll
        for (int it = 0; it < 2; ++it) {
            int vec = tid + it * 256;        // 0..511
            int row = vec >> 2;              // 0..127
            int c4  = (vec & 3) << 2;        // 0,4,8,12
            int gr  = m0 + row;
#if HAVE_ASYNC_LDS
            if (gr < M)
                async_copy_b128(A + (size_t)gr * lda + k0 + c4,
                                &Al[buf][row * LDST + c4]);
#else
            float4 v = make_float4(0.f, 0.f, 0.f, 0.f);
            if (gr < M) v = *(const float4*)(A + (size_t)gr * lda + k0 + c4);
            *(float4*)&Al[buf][row * LDST + c4] = v;
#endif
        }
        if (BT) {
#pragma unroll
            for (int it = 0; it < 2; ++it) {
                int vec = tid + it * 256;
                int row = vec >> 2;
                int c4  = (vec & 3) << 2;
                int gn  = n0 + row;
#if HAVE_ASYNC_LDS
                if (gn < Nb)
                    async_copy_b128(B + (size_t)gn * ldb + k0 + c4,
                                    &Bl[buf][row * LDST + c4]);
#else
                float4 v = make_float4(0.f, 0.f, 0.f, 0.f);
                if (gn < Nb) v = *(const float4*)(B + (size_t)gn * ldb + k0 + c4);
                *(float4*)&Bl[buf][row * LDST + c4] = v;
#endif
            }
        } else {
            // B is [K,N]: load row-slabs, store transposed (sync; async cannot transpose)
#pragma unroll
            for (int it = 0; it < 2; ++it) {
                int vec = tid + it * 256;
                int kk  = vec >> 5;              // 0..15
                int nn4 = (vec & 31) << 2;       // 0..124
                int gn  = n0 + nn4;
                float4 v;
                if (gn + 3 < Nb) {
                    v = *(const float4*)(B + (size_t)(k0 + kk) * ldb + gn);
                } else {
                    v.x = (gn + 0 < Nb) ? B[(size_t)(k0 + kk) * ldb + gn + 0] : 0.f;
                    v.y = (gn + 1 < Nb) ? B[(size_t)(k0 + kk) * ldb + gn + 1] : 0.f;
                    v.z = (gn + 2 < Nb) ? B[(size_t)(k0 + kk) * ldb + gn + 2] : 0.f;
                    v.w = (gn + 3 < Nb) ? B[(size_t)(k0 + kk) * ldb + gn + 3] : 0.f;
                }
                Bl[buf][(nn4 + 0) * LDST + kk] = v.x;
                Bl[buf][(nn4 + 1) * LDST + kk] = v.y;
                Bl[buf][(nn4 + 2) * LDST + kk] = v.z;
                Bl[buf][(nn4 + 3) * LDST + kk] = v.w;
            }
        }
    };

    v8f_t acc[2][4];
#pragma unroll
    for (int i = 0; i < 2; ++i)
#pragma unroll
        for (int j = 0; j < 4; ++j)
#pragma unroll
            for (int r = 0; r < 8; ++r) acc[i][j][r] = 0.0f;

    // prologue: fill buffer 0
    stage(0, 0);
    wait_async_all();
    __syncthreads();

    for (int t = 0; t < ntiles; ++t) {
        const int cur = t & 1;
        if (t + 1 < ntiles)
            stage(cur ^ 1, (t + 1) * KT);    // DMA next slab while we crunch this one

        const float* __restrict__ Ab = Al[cur];
        const float* __restrict__ Bb = Bl[cur];
        // A frag (ISA 7.12.2): lane16 -> M, VGPR/half -> K = k4 + 2*lh + {0,1}; B symmetric
#pragma unroll
        for (int k4 = 0; k4 < KT; k4 += 4) {
            const int kf = k4 + 2 * lh;
            v2f_t af[2], bf[4];
#pragma unroll
            for (int tm = 0; tm < 2; ++tm)
                af[tm] = *(const v2f_t*)&Ab[(wm * 32 + tm * 16 + l16) * LDST + kf];
#pragma unroll
            for (int tn = 0; tn < 4; ++tn)
                bf[tn] = *(const v2f_t*)&Bb[(wn * 64 + tn * 16 + l16) * LDST + kf];
#pragma unroll
            for (int tm = 0; tm < 2; ++tm)
#pragma unroll
                for (int tn = 0; tn < 4; ++tn)
                    acc[tm][tn] = __builtin_amdgcn_wmma_f32_16x16x4_f32(
                        false, af[tm], false, bf[tn], (short)0, acc[tm][tn], false, false);
        }
        wait_async_all();   // my async writes to the other buffer have landed
        __syncthreads();    // everyone done reading cur / writing nxt
    }

    // ---- epilogue ----
#pragma unroll
    for (int tm = 0; tm < 2; ++tm) {
#pragma unroll
        for (int tn = 0; tn < 4; ++tn) {
            int col = n0 + wn * 64 + tn * 16 + l16;
            if (col >= N) continue;
            float bs = bias ? bias[col] : 0.f;
            float rs = (mode == 1) ? rscale[col] : 1.f;
#pragma unroll
            for (int r = 0; r < 8; ++r) {
                int row = m0 + wm * 32 + tm * 16 + lh * 8 + r;
                if (row >= M) continue;
                float v = acc[tm][tn][r];
                float o;
                if (mode == 1)
                    o = 0.5f * v * rs + 0.5f * P[(size_t)row * ldp + col] + 0.5f * bs;
                else
                    o = v + bs;
                C[(size_t)row * ldc + col] = o;
            }
        }
    }
}

// ---------------- per-row argmax + softmax entropy ----------------
__global__ __launch_bounds__(256)
void argent_kernel(const float* __restrict__ logits, int ld, int ncols,
                   int* __restrict__ cls, float* __restrict__ ent, int offset)
{
    __shared__ float sv[256];
    __shared__ float sv2[256];
    __shared__ int   si[256];
    const int row = blockIdx.x;
    const int tid = threadIdx.x;
    const float* lp = logits + (size_t)row * ld;

    float m = -3.4e38f; int mi = 0x7fffffff;
    for (int c = tid; c < ncols; c += 256) {
        float v = lp[c];
        if (v > m || (v == m && c < mi)) { m = v; mi = c; }
    }
    sv[tid] = m; si[tid] = mi;
    __syncthreads();
    for (int s = 128; s > 0; s >>= 1) {
        if (tid < s) {
            float ov = sv[tid + s]; int oi = si[tid + s];
            if (ov > sv[tid] || (ov == sv[tid] && oi < si[tid])) { sv[tid] = ov; si[tid] = oi; }
        }
        __syncthreads();
    }
    const float gm = sv[0];
    const int gidx = si[0];
    __syncthreads();

    float s1 = 0.f, s2 = 0.f;
    for (int c = tid; c < ncols; c += 256) {
        float v = lp[c];
        float e = __expf(v - gm);
        s1 += e; s2 += e * v;
    }
    sv[tid] = s1; sv2[tid] = s2;
    __syncthreads();
    for (int s = 128; s > 0; s >>= 1) {
        if (tid < s) { sv[tid] += sv[tid + s]; sv2[tid] += sv2[tid + s]; }
        __syncthreads();
    }
    if (tid == 0) {
        float S = sv[0], T = sv2[0];
        float lse = gm + __logf(S);
        ent[offset + row] = lse - T / S;   // entropy = lse - E[logit]
        cls[offset + row] = gidx;
    }
}

// ---------------- per-class lowest-entropy rank < K mask ----------------
__global__ __launch_bounds__(256)
void rank_kernel(const int* __restrict__ cls, const float* __restrict__ ent,
                 float* __restrict__ maskf, int N, int K)
{
    __shared__ float se[2048];
    __shared__ int   sc[2048];
    const int tid = threadIdx.x;
    for (int j = tid; j < N; j += 256) { se[j] = ent[j]; sc[j] = cls[j]; }
    __syncthreads();
    int i = blockIdx.x * 256 + tid;
    if (i < N) {
        int c = sc[i]; float e = se[i]; int cnt = 0;
        for (int j = 0; j < N; ++j)
            cnt += (sc[j] == c && (se[j] < e || (se[j] == e && j < i))) ? 1 : 0;
        maskf[i] = (cnt < K) ? 1.f : 0.f;
    }
}

// ---------------- per-support row L2 scale (mask / max(norm,eps)) ----------------
__global__ __launch_bounds__(256)
void rowscale_kernel(const float* __restrict__ Wcls, const float* __restrict__ z,
                     int D, int nwarm, const float* __restrict__ maskf,
                     float* __restrict__ scale)
{
    __shared__ float red[256];
    const int i = blockIdx.x;
    const int tid = threadIdx.x;
    const float* rp = (i < nwarm) ? (Wcls + (size_t)i * D) : (z + (size_t)(i - nwarm) * D);
    float ss = 0.f;
    for (int d = tid; d < D; d += 256) { float v = rp[d]; ss += v * v; }
    red[tid] = ss;
    __syncthreads();
    for (int s = 128; s > 0; s >>= 1) {
        if (tid < s) red[tid] += red[tid + s];
        __syncthreads();
    }
    if (tid == 0)
        scale[i] = maskf[i] / fmaxf(sqrtf(red[0]), 1e-12f);
}

// ---------------- stable counting sort by class (deterministic) ----------------
__global__ void classsort_kernel(const int* __restrict__ cls, int N, int C,
                                 int* __restrict__ offs, int* __restrict__ order,
                                 int* __restrict__ cursor)
{
    if (threadIdx.x == 0 && blockIdx.x == 0) {
        for (int c = 0; c <= C; ++c) offs[c] = 0;
        for (int i = 0; i < N; ++i) offs[cls[i] + 1]++;
        for (int c = 0; c < C; ++c) offs[c + 1] += offs[c];
        for (int c = 0; c < C; ++c) cursor[c] = offs[c];
        for (int i = 0; i < N; ++i) order[cursor[cls[i]]++] = i;
    }
}

__global__ __launch_bounds__(256)
void zero_kernel(float* __restrict__ p, int n)
{
    int i = blockIdx.x * 256 + threadIdx.x;
    if (i < n) p[i] = 0.f;
}

// ---------------- weights[d][c] = sum over class-c supports of sn[i][d] ----------------
__global__ __launch_bounds__(256)
void weights_kernel(const float* __restrict__ Wcls, const float* __restrict__ z,
                    int D, int nwarm, const float* __restrict__ scale,
                    const int* __restrict__ offs, const int* __restrict__ order,
                    float* __restrict__ weights, int ldw)
{
    const int c = blockIdx.y;
    const int d = blockIdx.x * 256 + threadIdx.x;
    const int b = offs[c], e = offs[c + 1];
    float s = 0.f;
    for (int t = b; t < e; ++t) {
        int i = order[t];
        float sc = scale[i];
        if (sc != 0.f) {
            const float* rp = (i < nwarm) ? (Wcls + (size_t)i * D) : (z + (size_t)(i - nwarm) * D);
            s += rp[d] * sc;
        }
    }
    weights[(size_t)d * ldw + c] = s;
}

// ---------------- column rnorm of weights ----------------
__global__ __launch_bounds__(256)
void colnorm_kernel(const float* __restrict__ weights, int ldw, int D,
                    float* __restrict__ rscale)
{
    __shared__ float red[256];
    const int c = blockIdx.x;
    const int tid = threadIdx.x;
    float ss = 0.f;
    for (int d = tid; d < D; d += 256) { float v = weights[(size_t)d * ldw + c]; ss += v * v; }
    red[tid] = ss;
    __syncthreads();
    for (int s = 128; s > 0; s >>= 1) {
        if (tid < s) red[tid] += red[tid + s];
        __syncthreads();
    }
    if (tid == 0)
        rscale[c] = 1.f / fmaxf(sqrtf(red[0]), 1e-12f);
}

// ---------------- launcher ----------------
extern "C" void kernel_launch(void* const* d_in, const int* in_sizes, int n_in,
                              void* d_out, int out_size, void* d_ws, size_t ws_size,
                              hipStream_t stream)
{
    const float* x      = (const float*)d_in[0];   // [1024, 8192]
    const float* W_feat = (const float*)d_in[1];   // [2048, 8192]
    const float* b_feat = (const float*)d_in[2];   // [2048]
    const float* W_cls  = (const float*)d_in[3];   // [1000, 2048]
    const float* b_cls  = (const float*)d_in[4];   // [1000]
    float* out = (float*)d_out;                    // [1024, 1000]

    char* base = (char*)d_ws;
    size_t off = 0;
    auto carve = [&](size_t bytes) -> char* {
        char* p = base + off;
        off += (bytes + 255) & ~(size_t)255;
        return p;
    };
    float* z       = (float*)carve((size_t)B_ROWS * D_FEAT * 4);  // [1024,2048]
    float* p_log   = (float*)carve((size_t)B_ROWS * 1024 * 4);    // [1024,1024] (1000 valid)
    float* warm    = (float*)carve((size_t)1024 * 1024 * 4);      // [1000,1024] (1000 valid)
    float* weights = (float*)carve((size_t)D_FEAT * 1024 * 4);    // [2048,1024] (1000 valid)
    float* ent     = (float*)carve(2048 * 4);
    float* maskf   = (float*)carve(2048 * 4);
    float* scale   = (float*)carve(2048 * 4);
    float* rscale  = (float*)carve(1024 * 4);
    int*   cls     = (int*)carve(2048 * 4);
    int*   offs    = (int*)carve(1024 * 4);
    int*   order   = (int*)carve(2048 * 4);
    int*   cursor  = (int*)carve(1024 * 4);

    // 1) z = x @ W_feat^T + b_feat           (34.4 GFLOP, the big one)
    gemm_f32_wmma<true><<<dim3(D_FEAT / BN, B_ROWS / BM), 256, 0, stream>>>(
        x, IN_FEAT, W_feat, IN_FEAT, z, D_FEAT,
        B_ROWS, D_FEAT, IN_FEAT, D_FEAT, b_feat, 0, nullptr, nullptr, 0);

    // 2) p = z @ W_cls^T + b_cls  (ld 1024)
    gemm_f32_wmma<true><<<dim3(8, B_ROWS / BM), 256, 0, stream>>>(
        z, D_FEAT, W_cls, D_FEAT, p_log, 1024,
        B_ROWS, C_CLS, D_FEAT, C_CLS, b_cls, 0, nullptr, nullptr, 0);

    // 3) warm = W_cls @ W_cls^T + b_cls  (ld 1024)
    gemm_f32_wmma<true><<<dim3(8, 8), 256, 0, stream>>>(
        W_cls, D_FEAT, W_cls, D_FEAT, warm, 1024,
        C_CLS, C_CLS, D_FEAT, C_CLS, b_cls, 0, nullptr, nullptr, 0);

    // 4-5) pseudo-labels + entropies (warm supports first, then batch)
    argent_kernel<<<C_CLS, 256, 0, stream>>>(warm, 1024, C_CLS, cls, ent, 0);
    argent_kernel<<<B_ROWS, 256, 0, stream>>>(p_log, 1024, C_CLS, cls, ent, C_CLS);

    // 6) per-class top-FILTK lowest-entropy mask
    rank_kernel<<<(N_SUP + 255) / 256, 256, 0, stream>>>(cls, ent, maskf, N_SUP, FILTK);

    // 7) row L2 scale * mask
    rowscale_kernel<<<N_SUP, 256, 0, stream>>>(W_cls, z, D_FEAT, C_CLS, maskf, scale);

    // 8) deterministic stable sort of supports by class
    classsort_kernel<<<1, 32, 0, stream>>>(cls, N_SUP, C_CLS, offs, order, cursor);

    // 9) zero padded weights buffer, then class-bucketed reduction (= sn^T @ onehot)
    zero_kernel<<<(D_FEAT * 1024 + 255) / 256, 256, 0, stream>>>(weights, D_FEAT * 1024);
    weights_kernel<<<dim3(D_FEAT / 256, C_CLS), 256, 0, stream>>>(
        W_cls, z, D_FEAT, C_CLS, scale, offs, order, weights, 1024);

    // 10) per-column inverse norms
    colnorm_kernel<<<C_CLS, 256, 0, stream>>>(weights, 1024, D_FEAT, rscale);

    // 11) out = 0.5*(z @ weights)*rscale + 0.5*p + 0.5*b_cls
    gemm_f32_wmma<false><<<dim3(8, B_ROWS / BM), 256, 0, stream>>>(
        z, D_FEAT, weights, 1024, out, C_CLS,
        B_ROWS, C_CLS, D_FEAT, 1024, b_cls, 1, rscale, p_log, 1024);
}